// EncoderBlock_84739704750211
// MI455X (gfx1250) — compile-verified
//
#include <hip/hip_runtime.h>
#include <hip/hip_bf16.h>

// Problem constants (from the reference)
constexpr int BSZ  = 512;   // batch
constexpr int N_IN = 64;
constexpr int D_IN = 128;
constexpr int KSEL = 4;     // gathered rows per output channel
constexpr int N_OUT = 64;
constexpr int D_OUT = 128;
constexpr int NF = N_OUT * D_OUT;   // 8192 features
constexpr float EPS = 1e-5f;

typedef __attribute__((ext_vector_type(2))) float v2f;
typedef __attribute__((ext_vector_type(4))) float v4f;
typedef __attribute__((ext_vector_type(8))) float v8f;

// Matches the builtin's parameter pointee type exactly (from hipcc diagnostic):
//   __attribute__((__vector_size__(4 * sizeof(int)))) int
typedef int b128_t __attribute__((__vector_size__(4 * sizeof(int))));

// --- CDNA5 async global->LDS copy (ASYNCcnt-tracked), with safe fallback ----
#if defined(__has_builtin)
#  if __has_builtin(__builtin_amdgcn_global_load_async_to_lds_b128)
#    define HAVE_ASYNC_LDS 1
#  endif
#endif
#ifndef HAVE_ASYNC_LDS
#  define HAVE_ASYNC_LDS 0
#endif

__device__ __forceinline__ void wait_async_zero()
{
#if HAVE_ASYNC_LDS
    asm volatile("s_wait_asynccnt 0x0" ::: "memory");
#endif
}

// Stage one 64x128 f32 panel (8192 floats) global -> LDS, 256 threads.
__device__ __forceinline__ void stage_panel(const float* src, float* dstLds, int tid)
{
#if HAVE_ASYNC_LDS
    typedef __attribute__((address_space(1))) b128_t* gp_t;   // global int4*
    typedef __attribute__((address_space(3))) b128_t* lp_t;   // LDS    int4*
    gp_t g = (gp_t)src;
    lp_t l = (lp_t)dstLds;
    #pragma unroll
    for (int i = 0; i < 8; ++i) {   // 256 thr * 8 * 16B = 32 KB
        __builtin_amdgcn_global_load_async_to_lds_b128(
            g + i * 256 + tid, l + i * 256 + tid, /*offset=*/0, /*cpol=*/0);
    }
#else
    v4f*       dst = (v4f*)dstLds;
    const v4f* s   = (const v4f*)src;
    #pragma unroll
    for (int i = 0; i < 8; ++i) {
        dst[i * 256 + tid] = s[i * 256 + tid];
    }
#endif
}

// ---------------------------------------------------------------------------
// Kernel 1: y[b, o, e] = sum_{k,d} x[b, idx[o,k], d] * W[o,k,d,e] + bias[o,e]
// One block = (o, 128-row m-tile). 8 waves, each wave: 16 rows x 128 cols.
// K contraction (512) in 8 chunks of 64; W panels double-buffered in LDS via
// async global->LDS copies. f32 WMMA 16x16x4, full precision.
// ---------------------------------------------------------------------------
__global__ __launch_bounds__(256)
void gemm_bias_kernel(const float* __restrict__ x,
                      const float* __restrict__ W,
                      const float* __restrict__ bias,
                      const int*   __restrict__ idx,
                      float* __restrict__ y)
{
    __shared__ float Blds[2][64 * D_OUT];   // 2 x 32 KB ping-pong W panels

    const int o      = blockIdx.y;
    const int m_base = blockIdx.x * 128;
    const int tid    = threadIdx.x;
    const int wave   = tid >> 5;            // 0..7
    const int lane   = tid & 31;
    const int lane16 = lane & 15;
    const int khalf  = lane >> 4;           // 0: K={0,1}, 1: K={2,3} per WMMA step

    // Row of x this lane's A-fragment comes from (A layout: M = lane%16)
    const int mrow = m_base + wave * 16 + lane16;

    // gathered input rows for this output channel (uniform scalar loads)
    int nin[KSEL];
    #pragma unroll
    for (int k = 0; k < KSEL; ++k) nin[k] = idx[o * KSEL + k];

    const float* Wo = W + ((size_t)o * KSEL) * (size_t)(D_IN * D_OUT);
    // chunk c (0..7): k = c>>1, d-range = (c&1)*64 .. +63 ; panel is contiguous
    auto panel_ptr = [&](int c) -> const float* {
        return Wo + (size_t)(c >> 1) * (D_IN * D_OUT) + (size_t)(c & 1) * (64 * D_OUT);
    };

    v8f acc[8] = {};                        // 8 n-tiles of 16x16 f32 C/D

    // prologue: stage chunk 0 into buffer 0
    stage_panel(panel_ptr(0), &Blds[0][0], tid);
    wait_async_zero();
    __syncthreads();

    for (int c = 0; c < 8; ++c) {
        const int cur = c & 1;

        // prefetch next panel into the other buffer (overlaps with WMMA below)
        if (c + 1 < 8) stage_panel(panel_ptr(c + 1), &Blds[1 - cur][0], tid);

        const int dbase = (c & 1) * 64;
        const float* xrow =
            x + ((size_t)mrow * N_IN + nin[c >> 1]) * D_IN + dbase;
        const float* Bp = &Blds[cur][0];

        // 16 WMMA k-steps of 4 over this 64-wide chunk
        for (int ks = 0; ks < 16; ++ks) {
            const int kk = ks * 4 + khalf * 2;      // this lane's K pair
            v2f a = *(const v2f*)(xrow + kk);       // A frag from x row (8B aligned)
            #pragma unroll
            for (int nt = 0; nt < 8; ++nt) {
                const int n = nt * 16 + lane16;
                v2f b;
                b[0] = Bp[kk * D_OUT + n];          // B[kk  ][n]
                b[1] = Bp[(kk + 1) * D_OUT + n];    // B[kk+1][n]
                acc[nt] = __builtin_amdgcn_wmma_f32_16x16x4_f32(
                    /*neg_a=*/false, a, /*neg_b=*/false, b,
                    /*c_mod=*/(short)0, acc[nt],
                    /*reuse_a=*/false, /*reuse_b=*/false);
            }
        }

        if (c + 1 < 8) wait_async_zero();   // own async copies landed in LDS
        __syncthreads();                    // everyone done reading cur / writing next
    }

    // ---- epilogue: +bias, store. D layout: lanes 0-15 -> M=g, 16-31 -> M=g+8 ----
    const float* brow = bias + o * D_OUT;
    #pragma unroll
    for (int nt = 0; nt < 8; ++nt) {
        const int n  = nt * 16 + lane16;
        const float bv = brow[n];
        #pragma unroll
        for (int g = 0; g < 8; ++g) {
            const int m = m_base + wave * 16 + g + 8 * khalf;
            y[(size_t)m * NF + o * D_OUT + n] = acc[nt][g] + bv;
        }
    }
}

// ---------------------------------------------------------------------------
// Kernel 2: per-feature mean and rsqrt(var+eps) over the batch dim.
// Block = 256 threads = 64 features x 4 batch-groups. Grid = NF/64 = 128.
// stats[f] = mean, stats[NF+f] = rstd
// ---------------------------------------------------------------------------
__global__ __launch_bounds__(256)
void bn_stats_kernel(const float* __restrict__ y, float* __restrict__ stats)
{
    const int fl = threadIdx.x & 63;
    const int f  = blockIdx.x * 64 + fl;
    const int bg = threadIdx.x >> 6;   // 0..3

    float s = 0.f, ss = 0.f;
    for (int b = bg; b < BSZ; b += 4) {
        float v = y[(size_t)b * NF + f];
        s  += v;
        ss += v * v;
    }

    __shared__ float shs[4][64];
    __shared__ float shss[4][64];
    shs[bg][fl]  = s;
    shss[bg][fl] = ss;
    __syncthreads();

    if (bg == 0) {
        float S  = shs[0][fl]  + shs[1][fl]  + shs[2][fl]  + shs[3][fl];
        float SS = shss[0][fl] + shss[1][fl] + shss[2][fl] + shss[3][fl];
        float m   = S * (1.0f / BSZ);
        float var = SS * (1.0f / BSZ) - m * m;
        stats[f]      = m;
        stats[NF + f] = rsqrtf(var + EPS);
    }
}

// ---------------------------------------------------------------------------
// Kernel 3: in-place normalize + SiLU over y (float4 vectorized).
// Grid = BSZ*NF / (256*4) = 4096 blocks.
// ---------------------------------------------------------------------------
__global__ __launch_bounds__(256)
void bn_silu_kernel(float* __restrict__ y, const float* __restrict__ stats,
                    const float* __restrict__ gamma, const float* __restrict__ beta)
{
    const size_t i4 = ((size_t)blockIdx.x * 256 + threadIdx.x) * 4;
    const int f = (int)(i4 & (size_t)(NF - 1));     // NF multiple of 4, no row crossing

    v4f v  = *(const v4f*)(y + i4);
    v4f m  = *(const v4f*)(stats + f);
    v4f r  = *(const v4f*)(stats + NF + f);
    v4f g  = *(const v4f*)(gamma + f);
    v4f bt = *(const v4f*)(beta + f);

    v4f yn = (v - m) * r * g + bt;
    v4f out;
    #pragma unroll
    for (int c = 0; c < 4; ++c) {
        float t = yn[c];
        out[c] = t / (1.0f + __expf(-t));           // silu
    }
    *(v4f*)(y + i4) = out;
}

// ---------------------------------------------------------------------------
extern "C" void kernel_launch(void* const* d_in, const int* in_sizes, int n_in,
                              void* d_out, int out_size, void* d_ws, size_t ws_size,
                              hipStream_t stream)
{
    const float* x     = (const float*)d_in[0];
    const float* W     = (const float*)d_in[1];
    const float* bias  = (const float*)d_in[2];
    const float* gamma = (const float*)d_in[3];
    const float* beta  = (const float*)d_in[4];
    const int*   idx   = (const int*)  d_in[5];

    float* y     = (float*)d_out;     // (512, 64, 128) f32 — doubles as pre-BN buffer
    float* stats = (float*)d_ws;      // 2 * 8192 floats = 64 KB

    dim3 g1(BSZ / 128, N_OUT);        // (4, 64)
    gemm_bias_kernel<<<g1, 256, 0, stream>>>(x, W, bias, idx, y);
    bn_stats_kernel<<<NF / 64, 256, 0, stream>>>(y, stats);
    bn_silu_kernel<<<(BSZ * NF) / (256 * 4), 256, 0, stream>>>(y, stats, gamma, beta);
}